// EquivariantProductBasisBlock_41704132444229
// MI455X (gfx1250) — compile-verified
//
#include <hip/hip_runtime.h>

typedef __attribute__((ext_vector_type(2))) float v2f;
typedef __attribute__((ext_vector_type(8))) float v8f;
typedef __attribute__((ext_vector_type(4))) unsigned int u32x4;
typedef __attribute__((ext_vector_type(8))) int i32x8;
typedef __attribute__((ext_vector_type(4))) int i32x4;

#define Nn    2048
#define Cc    128
#define NIRRn 9
#define Ee    10

// ---------------- workspace layout (float offsets) -------------------------
// Upk0: [10][96][4]   (K-groups ks=0..9, cols n, kk pair-packed)  l=0, u3+u2
// Upk1: [10][256][4]                                             l=1, u3+u2
// A1  : [E][C][36]
// LW0p: [32][128][4]  packed lin_w0 for WMMA-B
// LW1p: [32][128][4]
// F0  : [2048][128]   f0 (node, channel)
// F1  : [6144][128]   f1 rows = (node*3 + m)
#define UPK0_OFF 0
#define UPK1_OFF 3840
#define A1_OFF   14080
#define LW0_OFF  60160
#define LW1_OFF  76544
#define F0_OFF   92928
#define F1_OFF   355072
#define WS_FLOATS 1141504   // ~4.6 MB

#define PREP_FLOATS F0_OFF
#define UPK_FLOATS  14080   // TDM tile: both packed U matrices

// ---------------------------------------------------------------------------
// Precompute: packed U matrices (u3 rows 0..35, u2 rows 36..38, row 39 = 0),
// A1 = u1*w1 per (e,c), and WMMA-packed lin_w matrices.
// ---------------------------------------------------------------------------
__global__ void mace_prep(const float* __restrict__ u3_0, const float* __restrict__ u2_0,
                          const float* __restrict__ u1_0, const float* __restrict__ w1_0,
                          const float* __restrict__ u3_1, const float* __restrict__ u2_1,
                          const float* __restrict__ u1_1, const float* __restrict__ w1_1,
                          const float* __restrict__ lw0, const float* __restrict__ lw1,
                          float* __restrict__ ws) {
  for (int idx = blockIdx.x * blockDim.x + threadIdx.x; idx < PREP_FLOATS;
       idx += gridDim.x * blockDim.x) {
    if (idx < UPK1_OFF) {                         // Upk0: [10][96][4]
      int ks = idx / 384, rem = idx % 384, n = rem >> 2, kk = rem & 3;
      float v = 0.f;
      if (n < 81) {
        int p = n / 9, q = n % 9;
        if (ks < 9)       v = u3_0[((p * 9 + q) * 9 + ks) * 4 + kk];   // d==0
        else if (kk < 3)  v = u2_0[(p * 9 + q) * 3 + kk];
      }
      ws[idx] = v;
    } else if (idx < A1_OFF) {                    // Upk1: [10][256][4]
      int t = idx - UPK1_OFF;
      int ks = t / 1024, rem = t % 1024, n = rem >> 2, kk = rem & 3;
      float v = 0.f;
      if (n < 243) {
        int d = n / 81, pq = n % 81, p = pq / 9, q = pq % 9;
        if (ks < 9)       v = u3_1[(((d * 9 + p) * 9 + q) * 9 + ks) * 4 + kk];
        else if (kk < 3)  v = u2_1[((d * 9 + p) * 9 + q) * 3 + kk];
      }
      ws[idx] = v;
    } else if (idx < LW0_OFF) {                   // A1[e][c][36]
      int t = idx - A1_OFF;
      int e = t / (Cc * 36), rem = t % (Cc * 36);
      int c = rem / 36, dp = rem % 36;
      float v = 0.f;
      if (dp < 9) {
        for (int k = 0; k < 2; ++k) v += u1_0[dp * 2 + k] * w1_0[(e * 2 + k) * Cc + c];
      } else {
        int tt = dp - 9, d = tt / 9, p = tt % 9;
        for (int k = 0; k < 2; ++k) v += u1_1[(d * 9 + p) * 2 + k] * w1_1[(e * 2 + k) * Cc + c];
      }
      ws[idx] = v;
    } else if (idx < LW1_OFF) {                   // LW0p: [32][128][4]
      int t = idx - LW0_OFF;
      int ks = t / 512, rem = t % 512, j = rem >> 2, kk = rem & 3;
      ws[idx] = lw0[(ks * 4 + kk) * Cc + j];
    } else {                                      // LW1p
      int t = idx - LW1_OFF;
      int ks = t / 512, rem = t % 512, j = rem >> 2, kk = rem & 3;
      ws[idx] = lw1[(ks * 4 + kk) * Cc + j];
    }
  }
}

// ---------------------------------------------------------------------------
// One WMMA tile pass (compile-time stride): C[128 x 16*NTN] = z[128x40] @ U
// followed by fused q-contraction into t2 (LDS float atomics).
// ---------------------------------------------------------------------------
template <int STRIDE, int NTN, int NVALID, int DPBASE>
__device__ __forceinline__ void tile_pass(const float* __restrict__ Up,
                                          const float* __restrict__ xsj,
                                          const float* __restrict__ w3Tj,
                                          const float* __restrict__ w2Tj,
                                          float* __restrict__ t2j,
                                          int wslot, int l, int half) {
  const v2f* Up2 = (const v2f*)Up;
  const v2f* w3p = (const v2f*)w3Tj;
  const v2f* w2p = (const v2f*)w2Tj;
  for (int t = wslot; t < 8 * NTN; t += 4) {
    const int mrow = t / NTN, ncol = t % NTN;
    const int n0 = ncol * 16, arow = mrow * 16 + l;
    const v2f wa  = w3p[arow * 2 + half];   // w3[e][kk..kk+1][arow]
    const v2f w2v = w2p[arow * 2 + half];   // w2 rows (K = 36..39, row 39 = 0)
    v8f acc = {};
#pragma unroll
    for (int ks = 0; ks < 9; ++ks) {        // z[c][i*4+k] = x[c][i]*w3[e][k][c]
      float xv = xsj[arow * 10 + ks];
      v2f a; a.x = xv * wa.x; a.y = xv * wa.y;
      v2f bb = Up2[(ks * STRIDE + n0 + l) * 2 + half];
      acc = __builtin_amdgcn_wmma_f32_16x16x4_f32(false, a, false, bb,
                                                  (short)0, acc, false, false);
    }
    {                                       // ks = 9: folded nu=2 (u2*w2) term
      v2f bb = Up2[(9 * STRIDE + n0 + l) * 2 + half];
      acc = __builtin_amdgcn_wmma_f32_16x16x4_f32(false, w2v, false, bb,
                                                  (short)0, acc, false, false);
    }
    const int n = n0 + l;
    if (n < NVALID) {
      const int q = n % 9, dp = DPBASE + n / 9;
#pragma unroll
      for (int r = 0; r < 8; ++r) {         // D VGPR r: M=r (lanes 0-15), M=8+r
        int crow = mrow * 16 + r + (half ? 8 : 0);
        atomicAdd(&t2j[crow * 37 + dp], acc[r] * xsj[crow * 10 + q]);
      }
    }
  }
}

// ---------------------------------------------------------------------------
// Main fused kernel: 2 nodes per block (amortizes U staging), 4 waves/node.
// Packed-U staging uses the Tensor Data Mover (one DMA issued by wave 0,
// tracked via TENSORcnt) while the other waves stage per-node data.
// ---------------------------------------------------------------------------
__launch_bounds__(256)
__global__ void mace_main(const float* __restrict__ x_in, const float* __restrict__ attrs,
                          const float* __restrict__ w3_0, const float* __restrict__ w2_0,
                          const float* __restrict__ w3_1, const float* __restrict__ w2_1,
                          float* __restrict__ ws) {
  extern __shared__ float smem[];
  float* u3s = smem;                 // 14080  (Upk0 | Upk1)
  float* xs  = u3s + 14080;          // 2 * [128][10]
  float* w3T = xs + 2560;            // 2 * 2 * [128][4]  (node, which)
  float* w2T = w3T + 2048;           // 2 * 2 * [128][4]
  float* t2  = w2T + 2048;           // 2 * [128][37]

  const int b0 = blockIdx.x * 2;
  const int tid = threadIdx.x;
  const int wave = tid >> 5, lane = tid & 31, half = lane >> 4, l = lane & 15;
  const int nodej = wave >> 2, wslot = wave & 3;

  int e[2];
  for (int j = 0; j < 2; ++j) {
    int ee = 0;
    for (int t = 0; t < Ee; ++t) if (attrs[(b0 + j) * Ee + t] > 0.5f) ee = t;
    e[j] = ee;
  }

  // ---- TDM: DMA both packed U matrices (56320 B) global -> LDS, wave 0 only.
  if (tid < 32) {
    const unsigned lds_base = (unsigned)(size_t)(void*)u3s;           // LDS byte addr
    const unsigned long long ga = (unsigned long long)(const void*)ws; // &ws[UPK0_OFF]
    u32x4 g0;
    g0.x = 1u;                                        // count=1 (valid), user mode
    g0.y = lds_base;                                  // D#.lds_addr
    g0.z = (unsigned)(ga & 0xFFFFFFFFu);              // global_addr[31:0]
    g0.w = (unsigned)((ga >> 32) & 0x1FFFFFFu)        // global_addr[56:32]
           | (2u << 30);                              // type = 2 ("image")
    i32x8 g1;
    g1[0] = (int)(2u << 16);                          // data_size = 4 B, mask = 0
    g1[1] = (int)((unsigned)UPK_FLOATS << 16);        // tensor_dim0[15:0]
    g1[2] = (int)(1u << 16);                          // tensor_dim0 hi=0 | tensor_dim1=1
    g1[3] = (int)((unsigned)UPK_FLOATS << 16);        // tile_dim0 = 14080
    g1[4] = 1;                                        // tile_dim1 = 1, tile_dim2 = 0
    g1[5] = UPK_FLOATS;                               // tensor_dim0_stride
    g1[6] = 0;
    g1[7] = 0;
    i32x4 gz4 = {0, 0, 0, 0};                         // groups 2/3 unused (<=2D)
    i32x8 gz8 = {0, 0, 0, 0, 0, 0, 0, 0};             // trailing group (clang-23 6-arg form)
    __builtin_amdgcn_tensor_load_to_lds(g0, g1, gz4, gz4, gz8, 0);
  }

  // ---- Meanwhile: per-node x / w3 / w2 / A1-initialized t2 (all waves)
  for (int j = 0; j < 2; ++j) {
    const int bb = b0 + j;
    for (int i = tid; i < Cc * NIRRn; i += 256) {
      int c = i / NIRRn, k = i % NIRRn;
      xs[j * 1280 + c * 10 + k] = x_in[(bb * Cc + c) * NIRRn + k];
    }
    for (int i = tid; i < 1024; i += 256) {     // (which, c, k)
      int which = i >> 9, rem = i & 511, c = rem >> 2, k = rem & 3;
      const float* w3g = which ? w3_1 : w3_0;
      const float* w2g = which ? w2_1 : w2_0;
      w3T[j * 1024 + which * 512 + c * 4 + k] = w3g[(e[j] * 4 + k) * Cc + c];
      w2T[j * 1024 + which * 512 + c * 4 + k] =
          (k < 3) ? w2g[(e[j] * 3 + k) * Cc + c] : 0.f;
    }
  }
  for (int i = tid; i < 2 * Cc * 36; i += 256) {
    int j = i / (Cc * 36), rem = i % (Cc * 36), c = rem / 36, dp = rem % 36;
    t2[j * 4736 + c * 37 + dp] = ws[A1_OFF + (e[j] * Cc + c) * 36 + dp];
  }

  if (tid < 32) __builtin_amdgcn_s_wait_tensorcnt(0);  // wave 0 waits its DMA
  __syncthreads();                                     // then all waves see LDS

  const float* xsj = xs + nodej * 1280;
  float* t2j = t2 + nodej * 4736;
  tile_pass<96, 6, 81, 0>(u3s, xsj, w3T + nodej * 1024, w2T + nodej * 1024,
                          t2j, wslot, l, half);
  tile_pass<256, 16, 243, 9>(u3s + UPK1_OFF, xsj, w3T + nodej * 1024 + 512,
                             w2T + nodej * 1024 + 512, t2j, wslot, l, half);
  __syncthreads();

  // p-contraction; emit f0/f1 in GEMM-ready layout for the linear kernel
  {
    const int j = tid >> 7, c = tid & 127;
    const float* xc = xs + j * 1280 + c * 10;
    const float* t2c = t2 + j * 4736 + c * 37;
    float s0 = 0.f, s1 = 0.f, s2 = 0.f, s3 = 0.f;
    for (int p = 0; p < 9; ++p) {
      float xv = xc[p];
      s0 += t2c[p] * xv;       s1 += t2c[9 + p] * xv;
      s2 += t2c[18 + p] * xv;  s3 += t2c[27 + p] * xv;
    }
    const int bb = b0 + j;
    ws[F0_OFF + bb * Cc + c] = s0;
    ws[F1_OFF + (bb * 3 + 0) * Cc + c] = s1;
    ws[F1_OFF + (bb * 3 + 1) * Cc + c] = s2;
    ws[F1_OFF + (bb * 3 + 2) * Cc + c] = s3;
  }
}

// ---------------------------------------------------------------------------
// o3.Linear as a batch WMMA GEMM: [2048x128]@lw0 and [6144x128]@lw1,
// scaled by 1/sqrt(C), + sc skip, scattered to the reference output layout.
// ---------------------------------------------------------------------------
__launch_bounds__(256)
__global__ void mace_linear(const float* __restrict__ ws, const float* __restrict__ sc,
                            float* __restrict__ out) {
  const int tid = threadIdx.x;
  const int wave = tid >> 5, lane = tid & 31, half = lane >> 4, l = lane & 15;
  const int g = blockIdx.x * 8 + wave;          // 0..4095 tiles

  const bool is0 = (g < 1024);
  const int gg = is0 ? g : (g - 1024);
  const int mrow = gg >> 3, ncol = gg & 7;
  const v2f* Ap2 = (const v2f*)(ws + (is0 ? F0_OFF : F1_OFF));
  const v2f* Bp2 = (const v2f*)(ws + (is0 ? LW0_OFF : LW1_OFF));

  const int n0 = ncol * 16, arow = mrow * 16 + l;
  __builtin_prefetch(&Bp2[(n0 + l) * 2], 0, 1);     // global_prefetch_b8

  v8f acc = {};
#pragma unroll 4
  for (int ks = 0; ks < 32; ++ks) {                 // K = 128
    v2f a  = Ap2[arow * 64 + ks * 2 + half];        // f[row][ks*4+kk..+1]
    v2f bb = Bp2[(ks * 128 + n0 + l) * 2 + half];   // lwpk[ks][n][kk..+1]
    acc = __builtin_amdgcn_wmma_f32_16x16x4_f32(false, a, false, bb,
                                                (short)0, acc, false, false);
  }

  const float inv = 0.08838834764831845f;           // 1/sqrt(128)
#pragma unroll
  for (int r = 0; r < 8; ++r) {
    const int row = mrow * 16 + r + (half ? 8 : 0);
    const int j = n0 + l;
    const float val = acc[r] * inv;
    size_t o;
    if (is0) {
      o = (size_t)row * 512 + j;                    // out[b, j]
    } else {
      const int b = row / 3, m = row - b * 3;
      o = (size_t)b * 512 + 128 + j * 3 + m;        // out[b, 128 + j*3 + m]
    }
    out[o] = val + sc[o];
  }
}

// ---------------------------------------------------------------------------
extern "C" void kernel_launch(void* const* d_in, const int* in_sizes, int n_in,
                              void* d_out, int out_size, void* d_ws, size_t ws_size,
                              hipStream_t stream) {
  const float* node_feats = (const float*)d_in[0];
  const float* node_attrs = (const float*)d_in[1];
  const float* sc   = (const float*)d_in[2];
  const float* u3_0 = (const float*)d_in[3];
  const float* u2_0 = (const float*)d_in[4];
  const float* u1_0 = (const float*)d_in[5];
  const float* w3_0 = (const float*)d_in[6];
  const float* w2_0 = (const float*)d_in[7];
  const float* w1_0 = (const float*)d_in[8];
  const float* u3_1 = (const float*)d_in[9];
  const float* u2_1 = (const float*)d_in[10];
  const float* u1_1 = (const float*)d_in[11];
  const float* w3_1 = (const float*)d_in[12];
  const float* w2_1 = (const float*)d_in[13];
  const float* w1_1 = (const float*)d_in[14];
  const float* lw0  = (const float*)d_in[15];
  const float* lw1  = (const float*)d_in[16];
  float* ws  = (float*)d_ws;     // needs WS_FLOATS*4 ~ 4.6 MB
  float* out = (float*)d_out;

  mace_prep<<<512, 256, 0, stream>>>(u3_0, u2_0, u1_0, w1_0,
                                     u3_1, u2_1, u1_1, w1_1, lw0, lw1, ws);

  const size_t lds = (size_t)(14080 + 2560 + 2048 + 2048 + 2 * 4736) * sizeof(float);
  mace_main<<<Nn / 2, 256, lds, stream>>>(node_feats, node_attrs,
                                          w3_0, w2_0, w3_1, w2_1, ws);

  mace_linear<<<512, 256, 0, stream>>>(ws, sc, out);
}